// ParamMemAdam_3204045603577
// MI455X (gfx1250) — compile-verified
//
#include <hip/hip_runtime.h>
#include <hip/hip_bf16.h>

typedef __attribute__((ext_vector_type(2))) float v2f;
typedef __attribute__((ext_vector_type(8))) float v8f;

// ---------------------------------------------------------------------------
// Kernel A: featurize the 64-entry vocab table.
// hs[b,t] depends only on seq[b,t] (per-token FF + LN), so we precompute
// table[v] = LN(emb[v] + FF(emb[v])) for v in [0,64).
// Uses V_WMMA_F32_16X16X4_F32 for the two GEMMs. 1 block, 4 waves,
// wave w handles vocab rows 16w..16w+15.
// ---------------------------------------------------------------------------
__global__ __launch_bounds__(128) void featurize_table(
    const float* __restrict__ emb,   // (64,64)
    const float* __restrict__ w1,    // (128,64)
    const float* __restrict__ b1,    // (128,)
    const float* __restrict__ w2,    // (64,128)
    const float* __restrict__ b2,    // (64,)
    const float* __restrict__ ln_g,  // (64,)
    const float* __restrict__ ln_b,  // (64,)
    float* __restrict__ table)       // (64,64)
{
    __shared__ float h1T[4][16][132];  // padded stride: avoid 64-bank conflicts
    __shared__ float xT[4][16][68];

    const int tid   = threadIdx.x;
    const int wave  = tid >> 5;
    const int lane  = tid & 31;
    const int m     = lane & 15;   // fragment row / col index
    const int khalf = lane >> 4;   // A/B: lanes 16-31 hold K=2,3
    const int rowbase = wave * 16; // vocab row base
    const int vrow  = rowbase + m;

    // GEMM1: h1 = relu(e @ W1^T + b1)   (16x64)x(64x128)
    for (int nt = 0; nt < 8; ++nt) {
        const int n = nt * 16 + m;
        const float bias = b1[n];
        v8f c = {bias, bias, bias, bias, bias, bias, bias, bias};
        for (int kt = 0; kt < 16; ++kt) {
            const int k0 = kt * 4 + khalf * 2;
            v2f a, bb;
            a.x  = emb[vrow * 64 + k0];
            a.y  = emb[vrow * 64 + k0 + 1];
            bb.x = w1[n * 64 + k0];        // B[k][n] = W1[n][k]
            bb.y = w1[n * 64 + k0 + 1];
            c = __builtin_amdgcn_wmma_f32_16x16x4_f32(false, a, false, bb,
                                                      (short)0, c, false, false);
        }
#pragma unroll
        for (int r = 0; r < 8; ++r) {
            const float v = c[r];
            h1T[wave][r + 8 * khalf][n] = v > 0.f ? v : 0.f;
        }
    }
    __syncthreads();

    // GEMM2: ff = h1 @ W2^T + b2 ; x = e + ff   (16x128)x(128x64)
    for (int nt = 0; nt < 4; ++nt) {
        const int n = nt * 16 + m;
        const float bias = b2[n];
        v8f c = {bias, bias, bias, bias, bias, bias, bias, bias};
        for (int kt = 0; kt < 32; ++kt) {
            const int k0 = kt * 4 + khalf * 2;
            v2f a, bb;
            a.x  = h1T[wave][m][k0];
            a.y  = h1T[wave][m][k0 + 1];
            bb.x = w2[n * 128 + k0];       // B[k][n] = W2[n][k]
            bb.y = w2[n * 128 + k0 + 1];
            c = __builtin_amdgcn_wmma_f32_16x16x4_f32(false, a, false, bb,
                                                      (short)0, c, false, false);
        }
#pragma unroll
        for (int r = 0; r < 8; ++r) {
            const int mr = r + 8 * khalf;
            xT[wave][mr][n] = c[r] + emb[(rowbase + mr) * 64 + n];
        }
    }
    __syncthreads();

    // LayerNorm per row: lane handles row (lane&15), half (lane>>4) of columns.
    {
        const int row = m;
        float s = 0.f, s2 = 0.f;
        for (int c0 = 0; c0 < 32; ++c0) {
            const float v = xT[wave][row][khalf * 32 + c0];
            s += v; s2 += v * v;
        }
        s  += __shfl_xor(s, 16, 32);
        s2 += __shfl_xor(s2, 16, 32);
        const float mu  = s * (1.f / 64.f);
        const float var = s2 * (1.f / 64.f) - mu * mu;
        const float inv = rsqrtf(var + 1e-5f);
        for (int c0 = 0; c0 < 32; ++c0) {
            const int col = khalf * 32 + c0;
            const float v = xT[wave][row][col];
            table[(rowbase + row) * 64 + col] = (v - mu) * inv * ln_g[col] + ln_b[col];
        }
    }
}

// ---------------------------------------------------------------------------
// Kernel B: per-batch-element sequential Adam chain (3069 steps).
// One workgroup per batch row; params shared in LDS; Adam moments in
// registers (17 params/thread). 4 barriers per step.
// Param layout (flat, 4192): fc1_w[0:2048] fc1_b[2048:2080]
//                            fc2_w[2080:4128] fc2_b[4128:4192]
// ---------------------------------------------------------------------------
#define NP 4192

__global__ __launch_bounds__(256) void ttt_adam(
    const int*   __restrict__ seq,      // (256,2048)
    const float* __restrict__ table_g,  // (64,64)
    const float* __restrict__ fc1_w, const float* __restrict__ fc1_b,
    const float* __restrict__ fc2_w, const float* __restrict__ fc2_b,
    float* __restrict__ ctx_out)        // (256,64)
{
    __shared__ float tbl[64 * 64];
    __shared__ int   seqb[2048];
    __shared__ float prm[NP];
    __shared__ float h1s[32];
    __shared__ float dpred[64];
    __shared__ float dh1s[32];

    const int b   = blockIdx.x;
    const int tid = threadIdx.x;

    // gfx1250 prefetch of this block's seq row
    __builtin_prefetch(&seq[b * 2048 + tid * 8], 0, 3);

    for (int i = tid; i < 4096; i += 256) tbl[i] = table_g[i];
    for (int i = tid; i < 2048; i += 256) seqb[i] = seq[b * 2048 + i];
    for (int i = tid; i < 2048; i += 256) prm[i] = fc1_w[i];
    if (tid < 32) prm[2048 + tid] = fc1_b[tid];
    for (int i = tid; i < 2048; i += 256) prm[2080 + i] = fc2_w[i];
    if (tid < 64) prm[4128 + tid] = fc2_b[tid];

    float m1r[17], m2r[17];
#pragma unroll
    for (int s = 0; s < 17; ++s) { m1r[s] = 0.f; m2r[s] = 0.f; }
    float b1p = 1.f, b2p = 1.f;
    __syncthreads();

    const int oh = tid >> 3, sub8 = tid & 7;   // groups of 8 lanes (32 outputs)
    const int o4 = tid >> 2, sub4 = tid & 3;   // groups of 4 lanes (64 outputs)

    int j = 0;
    for (int step = 0; step < 3069; ++step) {
        const int ik = seqb[2 * j];
        const int iv = seqb[2 * j + 1];
        const float* kv = &tbl[ik * 64];
        const float* vv = &tbl[iv * 64];

        // phase 1: h1 = relu(fc1_w @ k + fc1_b)
        {
            float acc = 0.f;
            const float* wrow = &prm[oh * 64 + sub8 * 8];
            const float* kk   = kv + sub8 * 8;
#pragma unroll
            for (int c = 0; c < 8; ++c) acc += wrow[c] * kk[c];
            acc += __shfl_xor(acc, 4, 32);
            acc += __shfl_xor(acc, 2, 32);
            acc += __shfl_xor(acc, 1, 32);
            if (sub8 == 0) {
                const float h = acc + prm[2048 + oh];
                h1s[oh] = h > 0.f ? h : 0.f;
            }
        }
        __syncthreads();

        // phase 2: dpred = 2/64 * (fc2_w @ h1 + fc2_b - v)
        {
            float acc = 0.f;
            const float* wrow = &prm[2080 + o4 * 32 + sub4 * 8];
#pragma unroll
            for (int c = 0; c < 8; ++c) acc += wrow[c] * h1s[sub4 * 8 + c];
            acc += __shfl_xor(acc, 2, 32);
            acc += __shfl_xor(acc, 1, 32);
            if (sub4 == 0)
                dpred[o4] = (acc + prm[4128 + o4] - vv[o4]) * (2.f / 64.f);
        }
        __syncthreads();

        // phase 3: dh1 = (fc2_w^T @ dpred) * relu'
        {
            float acc = 0.f;
#pragma unroll
            for (int c = 0; c < 8; ++c) {
                const int o = sub8 * 8 + c;
                acc += prm[2080 + o * 32 + oh] * dpred[o];
            }
            acc += __shfl_xor(acc, 4, 32);
            acc += __shfl_xor(acc, 2, 32);
            acc += __shfl_xor(acc, 1, 32);
            if (sub8 == 0) dh1s[oh] = (h1s[oh] > 0.f) ? acc : 0.f;
        }
        __syncthreads();

        // phase 4: fused gradient + Adam update
        b1p *= 0.9f; b2p *= 0.999f;
        const float ic1 = 1.f / (1.f - b1p);
        const float ic2 = 1.f / (1.f - b2p);
#pragma unroll
        for (int s = 0; s < 17; ++s) {
            const int p = tid + s * 256;
            if (p < NP) {
                float g;
                if (p < 2048)      g = dh1s[p >> 6] * kv[p & 63];
                else if (p < 2080) g = dh1s[p - 2048];
                else if (p < 4128) { const int q = p - 2080;
                                     g = dpred[q >> 5] * h1s[q & 31]; }
                else               g = dpred[p - 4128];
                const float m1 = 0.9f   * m1r[s] + 0.1f   * g;
                const float m2 = 0.999f * m2r[s] + 0.001f * g * g;
                m1r[s] = m1; m2r[s] = m2;
                prm[p] -= 0.05f * (m1 * ic1) / (sqrtf(m2 * ic2) + 1e-8f);
            }
        }
        __syncthreads();

        if (++j == 1023) j = 0;
    }

    // query forward through final params -> ctx
    {
        const int iq = seqb[2047];
        const float* qv = &tbl[iq * 64];
        float acc = 0.f;
        const float* wrow = &prm[oh * 64 + sub8 * 8];
#pragma unroll
        for (int c = 0; c < 8; ++c) acc += wrow[c] * qv[sub8 * 8 + c];
        acc += __shfl_xor(acc, 4, 32);
        acc += __shfl_xor(acc, 2, 32);
        acc += __shfl_xor(acc, 1, 32);
        if (sub8 == 0) {
            const float h = acc + prm[2048 + oh];
            h1s[oh] = h > 0.f ? h : 0.f;
        }
    }
    __syncthreads();
    {
        float acc = 0.f;
        const float* wrow = &prm[2080 + o4 * 32 + sub4 * 8];
#pragma unroll
        for (int c = 0; c < 8; ++c) acc += wrow[c] * h1s[sub4 * 8 + c];
        acc += __shfl_xor(acc, 2, 32);
        acc += __shfl_xor(acc, 1, 32);
        if (sub4 == 0) ctx_out[b * 64 + o4] = acc + prm[4128 + o4];
    }
}

// ---------------------------------------------------------------------------
// Kernel C: out = ctx @ out_w^T + out_b   (256x64)x(64x64), f32 WMMA.
// 4 blocks x 4 waves; each wave does a 16-row stripe.
// ---------------------------------------------------------------------------
__global__ __launch_bounds__(128) void out_proj(
    const float* __restrict__ ctx,    // (256,64)
    const float* __restrict__ out_w,  // (64,64)
    const float* __restrict__ out_b,  // (64,)
    float* __restrict__ out)          // (256,64)
{
    const int tid   = threadIdx.x;
    const int wave  = tid >> 5;
    const int lane  = tid & 31;
    const int m     = lane & 15;
    const int khalf = lane >> 4;
    const int rowbase = (blockIdx.x * 4 + wave) * 16;

    for (int nt = 0; nt < 4; ++nt) {
        const int n = nt * 16 + m;
        const float bias = out_b[n];
        v8f c = {bias, bias, bias, bias, bias, bias, bias, bias};
        for (int kt = 0; kt < 16; ++kt) {
            const int k0 = kt * 4 + khalf * 2;
            v2f a, bb;
            a.x  = ctx[(rowbase + m) * 64 + k0];
            a.y  = ctx[(rowbase + m) * 64 + k0 + 1];
            bb.x = out_w[n * 64 + k0];     // B[k][n] = out_w[n][k]
            bb.y = out_w[n * 64 + k0 + 1];
            c = __builtin_amdgcn_wmma_f32_16x16x4_f32(false, a, false, bb,
                                                      (short)0, c, false, false);
        }
#pragma unroll
        for (int r = 0; r < 8; ++r)
            out[(rowbase + r + 8 * khalf) * 64 + n] = c[r];
    }
}

// ---------------------------------------------------------------------------
extern "C" void kernel_launch(void* const* d_in, const int* in_sizes, int n_in,
                              void* d_out, int out_size, void* d_ws, size_t ws_size,
                              hipStream_t stream) {
    const int*   seq   = (const int*)  d_in[0];
    const float* emb   = (const float*)d_in[1];
    const float* ff_w1 = (const float*)d_in[2];
    const float* ff_b1 = (const float*)d_in[3];
    const float* ff_w2 = (const float*)d_in[4];
    const float* ff_b2 = (const float*)d_in[5];
    const float* ln_g  = (const float*)d_in[6];
    const float* ln_b  = (const float*)d_in[7];
    const float* fc1_w = (const float*)d_in[8];
    const float* fc1_b = (const float*)d_in[9];
    const float* fc2_w = (const float*)d_in[10];
    const float* fc2_b = (const float*)d_in[11];
    const float* out_w = (const float*)d_in[12];
    const float* out_b = (const float*)d_in[13];

    float* table = (float*)d_ws;          // 64*64 f32 = 16 KB
    float* ctx   = table + 64 * 64;       // 256*64 f32 = 64 KB
    float* out   = (float*)d_out;

    featurize_table<<<1, 128, 0, stream>>>(emb, ff_w1, ff_b1, ff_w2, ff_b2,
                                           ln_g, ln_b, table);
    ttt_adam<<<256, 256, 0, stream>>>(seq, table, fc1_w, fc1_b, fc2_w, fc2_b, ctx);
    out_proj<<<4, 128, 0, stream>>>(ctx, out_w, out_b, out);
}